// Residual_feature_79860621902105
// MI455X (gfx1250) — compile-verified
//
#include <hip/hip_runtime.h>
#include <stdint.h>

// ---------------------------------------------------------------------------
// Fused 3-stage depthwise conv (K1 -> K2 -> K3), fp32, 512x512 planes.
// One block computes a 64x16 output tile of one (n,c) plane.
// Pipeline: global --async b128 DMA--> LDS(in) -> LDS(edge) -> LDS(tex) -> out
// Per-stage zero-pad clipping of the reference is reproduced exactly.
// Roofline: fused single pass = 402 MB @ 23.3 TB/s ~= 17 us floor. Stages are
// register-blocked 4-wide so LDS read traffic (the next-binding resource after
// HBM) is halved vs the naive per-point form; stage-2 reads are exact-fit
// ds_load_b128 pairs. Interior blocks take clip-free code paths.
// ---------------------------------------------------------------------------

#define IMG       512
#define TX        64
#define TY        16
#define NTHREADS  256
#define LD_W      84                 // staged input width  [X0-8, X0+76), 4-aligned
#define LD_CHUNKS (LD_W / 4)         // 21 float4 chunks per row
#define IN_H      (TY + 12)          // 28 staged input rows [Y0-6, Y0+22)
#define E_W       (TX + 8)           // 72
#define E_H       (TY + 8)           // 24
#define E_G       (E_W / 4)          // 18 groups/row
#define T_W       (TX + 4)           // 68
#define T_H       (TY + 4)           // 20
#define T_G       (T_W / 4)          // 17 groups/row

typedef int v4i __attribute__((ext_vector_type(4)));
typedef __attribute__((address_space(1))) v4i* gptr_v4i;
typedef __attribute__((address_space(3))) v4i* lptr_v4i;
typedef __attribute__((address_space(1))) int* gptr_i32;
typedef __attribute__((address_space(3))) int* lptr_i32;

// ---- CDNA5 async global->LDS DMA (ASYNCcnt path) --------------------------
__device__ __forceinline__ void async_copy_f32x4(float* lds_dst, const float* gsrc) {
#if defined(__AMDGCN__) && __has_builtin(__builtin_amdgcn_global_load_async_to_lds_b128)
  __builtin_amdgcn_global_load_async_to_lds_b128(
      (gptr_v4i)gsrc, (lptr_v4i)lds_dst, /*imm offset=*/0, /*cpol=*/0);
#elif defined(__AMDGCN__) && __has_builtin(__builtin_amdgcn_global_load_async_to_lds_b32)
  __builtin_amdgcn_global_load_async_to_lds_b32((gptr_i32)gsrc + 0, (lptr_i32)lds_dst + 0, 0, 0);
  __builtin_amdgcn_global_load_async_to_lds_b32((gptr_i32)gsrc + 1, (lptr_i32)lds_dst + 1, 0, 0);
  __builtin_amdgcn_global_load_async_to_lds_b32((gptr_i32)gsrc + 2, (lptr_i32)lds_dst + 2, 0, 0);
  __builtin_amdgcn_global_load_async_to_lds_b32((gptr_i32)gsrc + 3, (lptr_i32)lds_dst + 3, 0, 0);
#elif defined(__AMDGCN__)
  unsigned lds_off = (unsigned)(uintptr_t)lds_dst;
  asm volatile("global_load_async_to_lds_b128 %0, %1, off"
               :: "v"(lds_off), "v"(gsrc) : "memory");
#else
  (void)lds_dst; (void)gsrc;
#endif
}

__device__ __forceinline__ void async_wait_all() {
#if defined(__AMDGCN__) && __has_builtin(__builtin_amdgcn_s_wait_asynccnt)
  __builtin_amdgcn_s_wait_asynccnt(0);
#elif defined(__AMDGCN__)
  asm volatile("s_wait_asynccnt 0x0" ::: "memory");
#endif
}

__device__ __forceinline__ void ld8(float* dst, const float* p) {  // 2x b128
  const float4 lo = *reinterpret_cast<const float4*>(p);
  const float4 hi = *reinterpret_cast<const float4*>(p + 4);
  dst[0] = lo.x; dst[1] = lo.y; dst[2] = lo.z; dst[3] = lo.w;
  dst[4] = hi.x; dst[5] = hi.y; dst[6] = hi.z; dst[7] = hi.w;
}

// ---- Stage 1: edge = x (*) K1, K1 core = [[-1,2,-1],[2,4,2],[-1,2,-1]] ----
// edge (r,c) <-> global (Y0-4+r, X0-4+c); s_in col 0 <-> gx = X0-8.
// 4 outputs/thread-task; 6 shared columns loaded once per row.
template <bool CLIP>
__device__ __forceinline__ void stage_edge(const float (*s_in)[LD_W],
                                           float (*s_edge)[E_W],
                                           int tid, int X0, int Y0) {
#pragma unroll
  for (int i = tid; i < E_H * E_G; i += NTHREADS) {   // 432 tasks -> 2 iters
    const int r  = i / E_G;
    const int c0 = (i - r * E_G) * 4;
    float s[6], b[6];
#pragma unroll
    for (int k = 0; k < 6; ++k) {
      s[k] = s_in[r + 1][c0 + 3 + k] + s_in[r + 3][c0 + 3 + k];  // top+bottom
      b[k] = s_in[r + 2][c0 + 3 + k];                            // middle
    }
    float v[4];
#pragma unroll
    for (int j = 0; j < 4; ++j) {
      const float u = s[j] + s[j + 2];               // corner pairs
      const float t = s[j + 1] + b[j] + b[j + 2];    // edge taps
      v[j] = 2.0f * t + 4.0f * b[j + 1] - u;
      if (CLIP) {
        const int gy = Y0 - 4 + r, gx = X0 - 4 + c0 + j;
        if (!((unsigned)gy < IMG && (unsigned)gx < IMG)) v[j] = 0.0f;
      }
    }
    *reinterpret_cast<float4*>(&s_edge[r][c0]) = float4{v[0], v[1], v[2], v[3]};
  }
}

// ---- Stage 2: tex = edge (*) K2 (doubly symmetric 5x5) --------------------
//   row0/4: [-1, 2,-2, 2,-1]  row1/3: [ 2,-6, 8,-6, 2]  row2: [-2, 8,-12, 8,-2]
// 4 outputs/thread-task need s_edge cols c0..c0+7 == exactly 2 aligned b128s.
template <bool CLIP>
__device__ __forceinline__ void stage_tex(const float (*s_edge)[E_W],
                                          float (*s_tex)[T_W],
                                          int tid, int X0, int Y0) {
#pragma unroll
  for (int i = tid; i < T_H * T_G; i += NTHREADS) {   // 340 tasks -> 2 iters
    const int r  = i / T_G;
    const int c0 = (i - r * T_G) * 4;
    float A[8], B[8], C[8], D[8], E[8];
    ld8(A, &s_edge[r + 0][c0]);
    ld8(E, &s_edge[r + 4][c0]);
    ld8(B, &s_edge[r + 1][c0]);
    ld8(D, &s_edge[r + 3][c0]);
    ld8(C, &s_edge[r + 2][c0]);
    float R0[8], R1[8];
#pragma unroll
    for (int k = 0; k < 8; ++k) { R0[k] = A[k] + E[k]; R1[k] = B[k] + D[k]; }
    float v[4];
#pragma unroll
    for (int j = 0; j < 4; ++j) {
      const float u0 = R0[j] + R0[j + 4], u1 = R0[j + 1] + R0[j + 3], u2 = R0[j + 2];
      const float p0 = R1[j] + R1[j + 4], p1 = R1[j + 1] + R1[j + 3], p2 = R1[j + 2];
      const float w0 = C[j] + C[j + 4],   w1 = C[j + 1] + C[j + 3],   w2 = C[j + 2];
      v[j] = -u0 + 2.0f * u1 - 2.0f * u2
           + 2.0f * p0 - 6.0f * p1 + 8.0f * p2
           - 2.0f * w0 + 8.0f * w1 - 12.0f * w2;
      if (CLIP) {
        const int gy = Y0 - 2 + r, gx = X0 - 2 + c0 + j;
        if (!((unsigned)gy < IMG && (unsigned)gx < IMG)) v[j] = 0.0f;
      }
    }
    *reinterpret_cast<float4*>(&s_tex[r][c0]) = float4{v[0], v[1], v[2], v[3]};
  }
}

// ---------------------------------------------------------------------------
__global__ __launch_bounds__(NTHREADS)
void fused_residual_feature_kernel(const float* __restrict__ x,
                                   float* __restrict__ out) {
  const int tid   = threadIdx.x;
  const int bx    = blockIdx.x;           // 0..7
  const int by    = blockIdx.y;           // 0..31
  const int plane = blockIdx.z;           // 0..191 (n*c)
  const int X0 = bx * TX;
  const int Y0 = by * TY;

  const float* __restrict__ src = x + (size_t)plane * IMG * IMG;

  __shared__ __align__(16) float s_in  [IN_H][LD_W];   //  9408 B
  __shared__ __align__(16) float s_edge[E_H][E_W];     //  6912 B
  __shared__ __align__(16) float s_tex [T_H][T_W];     //  5440 B

  const bool interior = (bx != 0) & (by != 0) &
                        (bx != (int)gridDim.x - 1) & (by != (int)gridDim.y - 1);

  // ---- Stage 0: async b128 DMA of [Y0-6,Y0+22) x [X0-8,X0+76) into LDS ----
  // Chunk starts are 0 mod 4 and IMG is 0 mod 4, so every 16B chunk is either
  // fully in-image (DMA) or fully out (zero-store); the sets are disjoint LDS
  // addresses, so no extra barrier is needed between them.
  {
    int r  = tid / LD_CHUNKS;             // 588 chunks total
    int c4 = tid - r * LD_CHUNKS;
#pragma unroll
    for (int it = 0; it < 3; ++it) {      // ceil(588/256)
      if (r < IN_H) {
        float* ldst = &s_in[r][c4 * 4];   // 16B-aligned: 84*4 = 21*16
        const int gy = Y0 - 6 + r;
        const int gx = X0 - 8 + c4 * 4;
        if (interior || ((unsigned)gy < IMG && (unsigned)gx < IMG)) {
          async_copy_f32x4(ldst, src + gy * IMG + gx);
        } else {
          *reinterpret_cast<float4*>(ldst) = float4{0.0f, 0.0f, 0.0f, 0.0f};
        }
      }
      c4 += NTHREADS - 12 * LD_CHUNKS;  r += 12;   // +4 chunks, +12 rows
      if (c4 >= LD_CHUNKS) { c4 -= LD_CHUNKS; r += 1; }
    }
    async_wait_all();
    __syncthreads();
  }

  // ---- Stages 1 & 2: block-uniform clip/no-clip specialization ------------
  if (interior) stage_edge<false>(s_in, s_edge, tid, X0, Y0);
  else          stage_edge<true >(s_in, s_edge, tid, X0, Y0);
  __syncthreads();

  if (interior) stage_tex<false>(s_edge, s_tex, tid, X0, Y0);
  else          stage_tex<true >(s_edge, s_tex, tid, X0, Y0);
  __syncthreads();

  // ---- Stage 3: residual = tex (*) K3, K3 = horizontal [1,-2,1] -----------
  // Each thread emits 4 consecutive x outputs -> one 16B store.
  {
    const int ox = (tid & 15) * 4;        // 0,4,...,60
    const int oy = tid >> 4;              // 0..15
    const float* tr = &s_tex[oy + 2][ox + 1];
    const float t0 = tr[0], t1 = tr[1], t2 = tr[2];
    const float t3 = tr[3], t4 = tr[4], t5 = tr[5];
    float4 o;
    o.x = t0 - 2.0f * t1 + t2;
    o.y = t1 - 2.0f * t2 + t3;
    o.z = t2 - 2.0f * t3 + t4;
    o.w = t3 - 2.0f * t4 + t5;
    float* dst = out + (size_t)plane * IMG * IMG + (size_t)(Y0 + oy) * IMG + (X0 + ox);
    *reinterpret_cast<float4*>(dst) = o;  // 16B-aligned: X0%64==0, ox%4==0
  }
}

// ---------------------------------------------------------------------------
extern "C" void kernel_launch(void* const* d_in, const int* in_sizes, int n_in,
                              void* d_out, int out_size, void* d_ws, size_t ws_size,
                              hipStream_t stream) {
  (void)in_sizes; (void)n_in; (void)out_size; (void)d_ws; (void)ws_size;
  const float* x = (const float*)d_in[0];
  float* out = (float*)d_out;
  dim3 grid(IMG / TX, IMG / TY, 64 * 3);   // 8 x 32 x 192 blocks
  dim3 block(NTHREADS);
  fused_residual_feature_kernel<<<grid, block, 0, stream>>>(x, out);
}